// SwinMLPBlock_85031762526272
// MI455X (gfx1250) — compile-verified
//
#include <hip/hip_runtime.h>
#include <hip/hip_bf16.h>

typedef float v2f __attribute__((ext_vector_type(2)));
typedef float v8f __attribute__((ext_vector_type(8)));

#define HEADS    16
#define HD       32      // head dim = 512/16
#define WS       8
#define WS2      64
#define NW       17      // windows per side on padded 136x136
#define IMG      128
#define CH       512
#define PADW     4       // ws - shift
#define LSTRIDE  67      // j-stride in LDS (67 mod 64 == 3 -> conflict-free)
#define NWAVES   4       // waves (= heads handled) per block

// One wave computes one (window, head): Y(64x32) = W_h(64x64) @ X(64x32) + b_h
// via 4x2 tiles of V_WMMA_F32_16X16X4_F32 over 16 K-steps.
__global__ __launch_bounds__(128) void swin_mlp_wmma(
    const float* __restrict__ x,     // (16,512,128,128)
    const float* __restrict__ w,     // (16,64,64)
    const float* __restrict__ bias,  // (16,64)
    float* __restrict__ out)         // (16,512,128,128)
{
    __shared__ float lds[NWAVES * HD * LSTRIDE];

    const int lane = threadIdx.x & 31;
    const int wave = threadIdx.x >> 5;

    const int hg   = blockIdx.x & 3;          // head group 0..3
    const int win  = blockIdx.x >> 2;         // window id
    const int head = hg * NWAVES + wave;

    const int ww = win % NW;
    const int wh = (win / NW) % NW;
    const int b  = win / (NW * NW);

    float* Xs = &lds[wave * HD * LSTRIDE];

    const int h0 = wh * WS - PADW;            // top row of window in image coords
    const int w0 = ww * WS - PADW;            // left col

    const float* xc = x + ((size_t)b * CH + (size_t)head * HD) * (IMG * IMG);

    // ---- Phase 1: gather window into LDS as Xs[d][j], j = r*8+col ----
    #pragma unroll 4
    for (int c = 0; c < HD; ++c) {
        #pragma unroll
        for (int jh = 0; jh < 2; ++jh) {
            int j   = jh * 32 + lane;
            int r   = j >> 3, col = j & 7;
            int hi  = h0 + r, wi = w0 + col;
            bool ok = ((unsigned)hi < IMG) && ((unsigned)wi < IMG);
            float v = ok ? xc[(size_t)c * (IMG * IMG) + hi * IMG + wi] : 0.0f;
            Xs[c * LSTRIDE + j] = v;
        }
    }

    __syncthreads();

    // ---- Phase 2: WMMA GEMM ----
    const int lmod  = lane & 15;
    const int lhalf = lane >> 4;

    const v8f zero = {0.f, 0.f, 0.f, 0.f, 0.f, 0.f, 0.f, 0.f};
    v8f acc[4][2];
    #pragma unroll
    for (int mt = 0; mt < 4; ++mt)
        #pragma unroll
        for (int nt = 0; nt < 2; ++nt)
            acc[mt][nt] = zero;

    const float* wh_ptr = w + (size_t)head * (WS2 * WS2);

    #pragma unroll
    for (int kk = 0; kk < 16; ++kk) {
        const int k0 = 4 * kk + 2 * lhalf;    // this lane's K pair

        // B tiles (K x N) from LDS: B[K=j][N=d] = Xs[d][j]
        v2f bt[2];
        #pragma unroll
        for (int nt = 0; nt < 2; ++nt) {
            int d = nt * 16 + lmod;
            bt[nt][0] = Xs[d * LSTRIDE + k0];
            bt[nt][1] = Xs[d * LSTRIDE + k0 + 1];
        }

        // A tiles (M x K) straight from global (w[h] is L2/L0 resident)
        #pragma unroll
        for (int mt = 0; mt < 4; ++mt) {
            v2f at = *(const v2f*)(wh_ptr + (mt * 16 + lmod) * WS2 + k0);
            #pragma unroll
            for (int nt = 0; nt < 2; ++nt) {
                acc[mt][nt] = __builtin_amdgcn_wmma_f32_16x16x4_f32(
                    false, at, false, bt[nt], (short)0, acc[mt][nt],
                    false, false);
            }
        }
    }

    __syncthreads();

    // ---- Phase 3: scatter accumulators into LDS as Ys[d][i] (reuse Xs) ----
    #pragma unroll
    for (int mt = 0; mt < 4; ++mt)
        #pragma unroll
        for (int nt = 0; nt < 2; ++nt)
            #pragma unroll
            for (int v = 0; v < 8; ++v)
                Xs[(nt * 16 + lmod) * LSTRIDE + mt * 16 + lhalf * 8 + v] =
                    acc[mt][nt][v];

    __syncthreads();

    // ---- Phase 4: coalesced store with bias (bias depends only on i=j) ----
    float* oc = out + ((size_t)b * CH + (size_t)head * HD) * (IMG * IMG);
    #pragma unroll
    for (int jh = 0; jh < 2; ++jh) {
        int j  = jh * 32 + lane;
        float bv = bias[head * WS2 + j];
        int r  = j >> 3, col = j & 7;
        int hi = h0 + r, wi = w0 + col;
        if (((unsigned)hi < IMG) && ((unsigned)wi < IMG)) {
            #pragma unroll 4
            for (int c = 0; c < HD; ++c) {
                oc[(size_t)c * (IMG * IMG) + hi * IMG + wi] =
                    Xs[c * LSTRIDE + j] + bv;
            }
        }
    }
}

extern "C" void kernel_launch(void* const* d_in, const int* in_sizes, int n_in,
                              void* d_out, int out_size, void* d_ws, size_t ws_size,
                              hipStream_t stream) {
    (void)in_sizes; (void)n_in; (void)out_size; (void)d_ws; (void)ws_size;
    const float* x    = (const float*)d_in[0];
    const float* w    = (const float*)d_in[1];
    const float* bias = (const float*)d_in[2];
    float* out        = (float*)d_out;

    const int nblocks = 16 * NW * NW * 4;   // batch * windows * head-groups
    hipLaunchKernelGGL(swin_mlp_wmma, dim3(nblocks), dim3(128), 0, stream,
                       x, w, bias, out);
}